// TransformerConv_29832842838838
// MI455X (gfx1250) — compile-verified
//
#include <hip/hip_runtime.h>
#include <math.h>

#define NNODES 10000
#define NEDGES 320000
#define NIN    256
#define NHEAD  8
#define NCH    32
#define DOUT   256   // NHEAD*NCH
#define SCALE  0.17677669529663687f   // 1/sqrt(32)

typedef float v2f __attribute__((ext_vector_type(2)));
typedef float v8f __attribute__((ext_vector_type(8)));

// IEEE-safe float atomic max: signed-int max for >=0, unsigned-int min for <0.
__device__ inline void atomic_max_f32(float* addr, float val) {
    if (!(val < 0.0f)) atomicMax((int*)addr, __float_as_int(val));
    else               atomicMin((unsigned int*)addr, __float_as_uint(val));
}

// ---------------------------------------------------------------------------
// Kernel 0: init segment-max to -inf and denom to 0
// ---------------------------------------------------------------------------
__global__ __launch_bounds__(256)
void init_softmax_state(float* __restrict__ m, float* __restrict__ denom) {
    int i = blockIdx.x * 256 + threadIdx.x;
    if (i < NNODES * NHEAD) {
        m[i]     = -INFINITY;
        denom[i] = 0.0f;
    }
}

// ---------------------------------------------------------------------------
// Kernel 1: fused 4-way GEMM via V_WMMA_F32_16X16X4_F32.
// One wave computes one 16x16 output tile; K-loop of 64 wmma (K=4) steps.
// Skip projection is written straight into d_out (residual base).
// ---------------------------------------------------------------------------
__global__ __launch_bounds__(256)
void gemm_qkvs_wmma(const float* __restrict__ X,
                    const float* __restrict__ Wq, const float* __restrict__ bq,
                    const float* __restrict__ Wk, const float* __restrict__ bk,
                    const float* __restrict__ Wv, const float* __restrict__ bv,
                    const float* __restrict__ Ws, const float* __restrict__ bs,
                    float* __restrict__ q, float* __restrict__ k,
                    float* __restrict__ v, float* __restrict__ outskip)
{
    const int TILES_PER_W = (NNODES / 16) * (DOUT / 16);   // 625 * 16 = 10000
    const int wid = blockIdx.x * 8 + (threadIdx.x >> 5);
    if (wid >= 4 * TILES_PER_W) return;

    const int widx = wid / TILES_PER_W;
    const int t    = wid % TILES_PER_W;
    const int m0   = (t / (DOUT / 16)) * 16;
    const int n0   = (t % (DOUT / 16)) * 16;

    const float* W; const float* bias; float* out;
    if      (widx == 0) { W = Wq; bias = bq; out = q;       }
    else if (widx == 1) { W = Wk; bias = bk; out = k;       }
    else if (widx == 2) { W = Wv; bias = bv; out = v;       }
    else                { W = Ws; bias = bs; out = outskip; }

    const int lane = threadIdx.x & 31;
    const int row  = lane & 15;   // M (for A) / N (for B,C)
    const int kh   = lane >> 4;   // which K pair (A/B), which M half (C/D)
    const int col  = n0 + row;

    // C/D layout: VGPR r, lanes 0-15 -> M=r, lanes 16-31 -> M=r+8; N = lane&15.
    // Bias is per-column -> broadcast into all 8 accumulator elements.
    v8f c;
    const float bcol = bias[col];
    #pragma unroll
    for (int r = 0; r < 8; ++r) c[r] = bcol;

    const float* xrow = X + (size_t)(m0 + row) * NIN;
    #pragma unroll 4
    for (int k0 = 0; k0 < NIN; k0 += 4) {
        // A 16x4: lane<16 holds K = {0,1}, lane>=16 holds K = {2,3} (contiguous -> b64 load)
        v2f a = *(const v2f*)(xrow + k0 + 2 * kh);
        // B 4x16 (row-major W[k][n]): VGPR0 rows K=0/2, VGPR1 rows K=1/3
        v2f b;
        b.x = W[(size_t)(k0 + 2 * kh)     * DOUT + col];
        b.y = W[(size_t)(k0 + 2 * kh + 1) * DOUT + col];
        c = __builtin_amdgcn_wmma_f32_16x16x4_f32(false, a, false, b,
                                                  (short)0, c, false, false);
    }

    #pragma unroll
    for (int r = 0; r < 8; ++r)
        out[(size_t)(m0 + r + 8 * kh) * DOUT + col] = c[r];
}

// ---------------------------------------------------------------------------
// Kernel 2: per-edge attention logits + segment max.
// One wave per edge; lane = channel within a head; 8 heads sequential.
// ---------------------------------------------------------------------------
__global__ __launch_bounds__(256)
void edge_scores(const float* __restrict__ q, const float* __restrict__ k,
                 const int* __restrict__ src, const int* __restrict__ dst,
                 float* __restrict__ score, float* __restrict__ m)
{
    const int e = blockIdx.x * 8 + (threadIdx.x >> 5);
    if (e >= NEDGES) return;
    const int lane = threadIdx.x & 31;
    const int s = src[e], d = dst[e];
    const float* qrow = q + (size_t)d * DOUT;
    const float* krow = k + (size_t)s * DOUT;

    #pragma unroll
    for (int h = 0; h < NHEAD; ++h) {
        float p = qrow[h * NCH + lane] * krow[h * NCH + lane];
        p += __shfl_xor(p, 16, 32);
        p += __shfl_xor(p,  8, 32);
        p += __shfl_xor(p,  4, 32);
        p += __shfl_xor(p,  2, 32);
        p += __shfl_xor(p,  1, 32);
        if (lane == 0) {
            float sc = p * SCALE;
            score[(size_t)e * NHEAD + h] = sc;
            atomic_max_f32(&m[d * NHEAD + h], sc);
        }
    }
}

// ---------------------------------------------------------------------------
// Kernel 3: exp(score - max) in place + segment sum of exponentials.
// One thread per (edge, head).
// ---------------------------------------------------------------------------
__global__ __launch_bounds__(256)
void edge_exp_denom(float* __restrict__ score, const float* __restrict__ m,
                    float* __restrict__ denom, const int* __restrict__ dst)
{
    int i = blockIdx.x * 256 + threadIdx.x;
    if (i >= NEDGES * NHEAD) return;
    int e = i >> 3;
    int h = i & 7;
    int d = dst[e];
    float ex = __expf(score[i] - m[d * NHEAD + h]);
    score[i] = ex;                    // reuse buffer: score -> exp(score - max)
    atomicAdd(&denom[d * NHEAD + h], ex);
}

// ---------------------------------------------------------------------------
// Kernel 4: alpha-weighted scatter-add of v[src] into out[dst] (on top of skip).
// One wave per edge; 256 f32 atomics per edge (L2-resident).
// ---------------------------------------------------------------------------
__global__ __launch_bounds__(256)
void edge_scatter_v(const float* __restrict__ ex, const float* __restrict__ denom,
                    const float* __restrict__ v,
                    const int* __restrict__ src, const int* __restrict__ dst,
                    float* __restrict__ out)
{
    const int e = blockIdx.x * 8 + (threadIdx.x >> 5);
    if (e >= NEDGES) return;
    const int lane = threadIdx.x & 31;
    const int s = src[e], d = dst[e];
    const float* vrow = v + (size_t)s * DOUT;
    float* orow = out + (size_t)d * DOUT;

    #pragma unroll
    for (int h = 0; h < NHEAD; ++h) {
        float alpha = ex[(size_t)e * NHEAD + h] / denom[d * NHEAD + h];
        atomicAdd(&orow[h * NCH + lane], alpha * vrow[h * NCH + lane]);
    }
}

// ---------------------------------------------------------------------------
extern "C" void kernel_launch(void* const* d_in, const int* in_sizes, int n_in,
                              void* d_out, int out_size, void* d_ws, size_t ws_size,
                              hipStream_t stream)
{
    const float* x     = (const float*)d_in[0];
    const int*   eidx  = (const int*)  d_in[1];
    const float* Wq    = (const float*)d_in[2];
    const float* bq    = (const float*)d_in[3];
    const float* Wk    = (const float*)d_in[4];
    const float* bk    = (const float*)d_in[5];
    const float* Wv    = (const float*)d_in[6];
    const float* bv    = (const float*)d_in[7];
    const float* Wsk   = (const float*)d_in[8];
    const float* bsk   = (const float*)d_in[9];
    float*       out   = (float*)d_out;

    const int* src = eidx;            // edge_index[0]
    const int* dst = eidx + NEDGES;   // edge_index[1]

    // Workspace layout (floats)
    float* ws    = (float*)d_ws;
    const size_t NQ = (size_t)NNODES * DOUT;   // 2.56M floats
    float* q     = ws;
    float* k     = q + NQ;
    float* v     = k + NQ;
    float* score = v + NQ;                      // E*H, reused as exp buffer
    float* m     = score + (size_t)NEDGES * NHEAD;
    float* denom = m + (size_t)NNODES * NHEAD;

    // 0) init softmax running state
    init_softmax_state<<<(NNODES * NHEAD + 255) / 256, 256, 0, stream>>>(m, denom);

    // 1) q,k,v projections + skip projection straight into d_out
    {
        const int nwaves = 4 * (NNODES / 16) * (DOUT / 16);   // 40000
        gemm_qkvs_wmma<<<(nwaves + 7) / 8, 256, 0, stream>>>(
            x, Wq, bq, Wk, bk, Wv, bv, Wsk, bsk, q, k, v, out);
    }

    // 2) attention logits + segment max
    edge_scores<<<(NEDGES + 7) / 8, 256, 0, stream>>>(q, k, src, dst, score, m);

    // 3) exp + segment sum
    edge_exp_denom<<<(NEDGES * NHEAD + 255) / 256, 256, 0, stream>>>(score, m, denom, dst);

    // 4) weighted scatter-add onto skip
    edge_scatter_v<<<(NEDGES + 7) / 8, 256, 0, stream>>>(score, denom, v, src, dst, out);
}